// PiecewiseActivation_9646496546830
// MI455X (gfx1250) — compile-verified
//
#include <hip/hip_runtime.h>

// Piecewise GELU (tanh approximation with +/-3 clamps) over fp32.
// Memory-bound: 8 B/element, ~536 MB total -> ~23 us floor at 23.3 TB/s.
// Strategy: 128-bit non-temporal streaming loads/stores (TH_NT: 512 MB
// working set >> 192 MB L2, touched once), hardware v_tanh_f32 where
// available, wave32-friendly 256-thread blocks, 4 float4 per thread with a
// branchless fast path so all 4 b128 loads are in flight simultaneously.

#define SQRT_2_OVER_PI 0.7978845608028654f
#define GELU_COEF      0.044715f
#define CLAMP_V        3.0f

// Native clang vector type: required by __builtin_nontemporal_{load,store},
// same 16-byte layout as 4 packed floats -> global_{load,store}_b128.
typedef __attribute__((ext_vector_type(4))) float v4f;

__device__ __forceinline__ float fast_tanh(float x) {
#if __has_builtin(__builtin_amdgcn_tanhf)
    return __builtin_amdgcn_tanhf(x);
#elif __has_builtin(__builtin_amdgcn_tanh_f32)
    return __builtin_amdgcn_tanh_f32(x);
#else
    return tanhf(x);
#endif
}

__device__ __forceinline__ float gelu_piecewise(float x) {
    float x3    = x * x * x;
    float inner = SQRT_2_OVER_PI * __builtin_fmaf(GELU_COEF, x3, x);
    float t     = fast_tanh(inner);
    float mid   = 0.5f * x * (1.0f + t);
    float r     = (x <= -CLAMP_V) ? 0.0f : mid;
    r           = (x >=  CLAMP_V) ? x    : r;
    return r;
}

__device__ __forceinline__ v4f gelu_vec4(v4f v) {
    v4f r;
    r.x = gelu_piecewise(v.x);
    r.y = gelu_piecewise(v.y);
    r.z = gelu_piecewise(v.z);
    r.w = gelu_piecewise(v.w);
    return r;
}

// 4 float4 per thread, block-strided so consecutive lanes touch consecutive
// 16B chunks (fully coalesced b128 traffic).
#define VPT 4

__global__ __launch_bounds__(256) void gelu_kernel(const float* __restrict__ in,
                                                   float* __restrict__ out,
                                                   long long nvec,   // number of float4s
                                                   long long ntail)  // leftover scalars
{
    const v4f* __restrict__ vin  = reinterpret_cast<const v4f*>(in);
    v4f* __restrict__       vout = reinterpret_cast<v4f*>(out);

    const long long stride    = blockDim.x;                       // 256
    const long long blockBase = (long long)blockIdx.x * (stride * VPT);
    const long long base      = blockBase + threadIdx.x;

    if (blockBase + stride * VPT <= nvec) {
        // -------- Fast path: whole block in bounds (all but last block). --------
        // Issue all 4 NT b128 loads before any use -> 4 outstanding loads/wave.
        v4f v[VPT];
#pragma unroll
        for (int k = 0; k < VPT; ++k)
            v[k] = __builtin_nontemporal_load(vin + base + (long long)k * stride);

        v4f r[VPT];
#pragma unroll
        for (int k = 0; k < VPT; ++k)
            r[k] = gelu_vec4(v[k]);

#pragma unroll
        for (int k = 0; k < VPT; ++k)
            __builtin_nontemporal_store(r[k], vout + base + (long long)k * stride);
    } else {
        // -------- Guarded path: final partial block only. --------
#pragma unroll
        for (int k = 0; k < VPT; ++k) {
            long long i = base + (long long)k * stride;
            if (i < nvec) {
                v4f v = __builtin_nontemporal_load(vin + i);
                __builtin_nontemporal_store(gelu_vec4(v), vout + i);
            }
        }
        // Scalar tail (n % 4 != 0). Zero for this problem; kept for safety.
        if (blockIdx.x == 0 && threadIdx.x < ntail) {
            long long i = nvec * 4 + threadIdx.x;
            out[i] = gelu_piecewise(in[i]);
        }
    }
}

extern "C" void kernel_launch(void* const* d_in, const int* in_sizes, int n_in,
                              void* d_out, int out_size, void* d_ws, size_t ws_size,
                              hipStream_t stream) {
    (void)n_in; (void)d_ws; (void)ws_size; (void)out_size;

    const float* in  = reinterpret_cast<const float*>(d_in[0]);
    float*       out = reinterpret_cast<float*>(d_out);

    long long n     = (long long)in_sizes[0];
    long long nvec  = n >> 2;         // float4 count
    long long ntail = n & 3LL;

    const int threads = 256;          // 8 wave32 waves per block
    long long vec_per_block = (long long)threads * VPT;
    long long blocks = (nvec + vec_per_block - 1) / vec_per_block;
    if (blocks < 1) blocks = 1;
    if (ntail > 0 && blocks < 1) blocks = 1;   // tail handled by block 0

    gelu_kernel<<<dim3((unsigned)blocks), dim3(threads), 0, stream>>>(in, out, nvec, ntail);
}